// AttentiveAggregation_6468220747834
// MI455X (gfx1250) — compile-verified
//
#include <hip/hip_runtime.h>

typedef float v2f __attribute__((ext_vector_type(2)));
typedef float v4f __attribute__((ext_vector_type(4)));
typedef float v8f __attribute__((ext_vector_type(8)));

#define DCOLS 128            // feature dim (fixed by reference)
#define TILE_ROWS 16         // rows per WMMA K-tile group
#define TILES_PER_CHUNK 8    // 128 rows per wave-chunk
#define CHUNK_ROWS 128
#define LDS_STRIDE 132       // 128 + 4 pad: kills bank conflicts on row-indexed reads
#define MAX_SLOTS 16         // M dimension of WMMA = live graph slots per wave
#define NCT 8                // 128 cols / 16 per WMMA N-tile

// Issue one 16-row H tile as 16 async direct-to-LDS b128 copies (ASYNCcnt).
// Each lane moves 16B per row: H[row*128 + lane*4 ..] -> hlds[r*132 + lane*4 ..]
__device__ __forceinline__ void issue_tile_async(const float* __restrict__ H,
                                                 long long row0, unsigned lds_lane_base,
                                                 int lane, long long V)
{
  const float* gp = H + row0 * DCOLS + lane * 4;
  if (row0 + TILE_ROWS <= V) {
    #pragma unroll
    for (int r = 0; r < TILE_ROWS; ++r) {
      unsigned la = lds_lane_base + (unsigned)(r * LDS_STRIDE * 4);
      unsigned long long ga = (unsigned long long)(gp + r * DCOLS);
      asm volatile("global_load_async_to_lds_b128 %0, %1, off"
                   :: "v"(la), "v"(ga) : "memory");
    }
  } else {
    #pragma unroll
    for (int r = 0; r < TILE_ROWS; ++r) {
      if (row0 + r < V) {   // uniform per-row guard (tail tile only)
        unsigned la = lds_lane_base + (unsigned)(r * LDS_STRIDE * 4);
        unsigned long long ga = (unsigned long long)(gp + r * DCOLS);
        asm volatile("global_load_async_to_lds_b128 %0, %1, off"
                     :: "v"(la), "v"(ga) : "memory");
      }
    }
  }
}

// Fused single pass over H: logit = exp(H.w+b); Z[g] += logit (atomic);
// S[g,:] += logit * H[v,:] on the matrix unit (v_wmma_f32_16x16x4_f32) with a
// 16-graph-slot accumulator window per wave (batch is sorted), flushed with
// fp32 atomics on window overflow / chunk end. H tiles stream through LDS via
// double-buffered global_load_async_to_lds_b128.
__global__ __launch_bounds__(32) void attn_pool_fused(
    const float* __restrict__ H, const int* __restrict__ batch,
    const float* __restrict__ w, const float* __restrict__ b,
    float* __restrict__ out, float* __restrict__ Z, int V)
{
  __shared__ float hlds[2][TILE_ROWS * LDS_STRIDE];   // 2 x 8448 B
  __shared__ float wlds[DCOLS];
  __shared__ float logit_lds[TILE_ROWS];
  __shared__ int   slot_lds[TILE_ROWS];
  __shared__ int   active_ids[MAX_SLOTS];

  const int lane   = threadIdx.x;     // wave32
  const int lane16 = lane & 15;
  const int half   = lane >> 4;
  const long long chunk_start = (long long)blockIdx.x * CHUNK_ROWS;
  if (chunk_start >= V) return;
  const long long chunk_end = (chunk_start + CHUNK_ROWS < V) ? (chunk_start + CHUNK_ROWS) : V;
  const int ntiles = (int)((chunk_end - chunk_start + TILE_ROWS - 1) / TILE_ROWS);

  *(v4f*)(wlds + lane * 4) = *(const v4f*)(w + lane * 4);
  const float bias = b[0];

  const unsigned lds0 = (unsigned)(unsigned long long)(&hlds[0][0]) + (unsigned)(lane * 16);
  const unsigned lds1 = (unsigned)(unsigned long long)(&hlds[1][0]) + (unsigned)(lane * 16);

  v8f acc[NCT];
  #pragma unroll
  for (int ct = 0; ct < NCT; ++ct)
    #pragma unroll
    for (int j = 0; j < 8; ++j) acc[ct][j] = 0.0f;

  int na = 0;                       // live slot count (wave-uniform register)

  issue_tile_async(H, chunk_start, lds0, lane, V);   // prologue: tile 0 in flight
  __syncthreads();

  for (int t = 0; t < ntiles; ++t) {
    const long long row0 = chunk_start + (long long)t * TILE_ROWS;
    float* hb = &hlds[t & 1][0];

    // ---- issue next tile into the other buffer, then wait for current ----
    bool next_full = false;
    if (t + 1 < ntiles) {
      long long nrow0 = row0 + TILE_ROWS;
      next_full = (nrow0 + TILE_ROWS <= V);
      issue_tile_async(H, nrow0, (t & 1) ? lds0 : lds1, lane, V);
    }
    if ((t + 1 < ntiles) && next_full)
      asm volatile("s_wait_asynccnt 0x10" ::: "memory");  // <=16 left => tile t done
    else
      asm volatile("s_wait_asynccnt 0x0" ::: "memory");

    // tail tile: zero-fill rows beyond V (uniform branch; never taken for V%16==0)
    if (row0 + TILE_ROWS > V) {
      #pragma unroll
      for (int r = 0; r < TILE_ROWS; ++r)
        if (row0 + r >= V) *(v4f*)(hb + r * LDS_STRIDE + lane * 4) = (v4f){0.f,0.f,0.f,0.f};
      __syncthreads();
    }

    // ---- logits: lane handles row (lane&15), column half (lane>>4) ----
    const long long myrow = row0 + lane16;
    const bool valid = (lane < TILE_ROWS) && (myrow < V);
    int myid = valid ? batch[myrow] : -1;

    {
      const float* hr = hb + lane16 * LDS_STRIDE + half * 64;
      const float* wr = wlds + half * 64;
      float p = 0.0f;
      #pragma unroll
      for (int c = 0; c < 64; c += 4) {
        v4f hv = *(const v4f*)(hr + c);
        v4f wv = *(const v4f*)(wr + c);
        p = fmaf(hv.x, wv.x, p);
        p = fmaf(hv.y, wv.y, p);
        p = fmaf(hv.z, wv.z, p);
        p = fmaf(hv.w, wv.w, p);
      }
      p += __shfl_xor(p, 16);
      float lg = 0.0f;
      if (valid) {
        lg = __expf(p + bias);
        atomicAdd(&Z[myid], lg);        // per-graph normalizer
      }
      if (lane < TILE_ROWS) logit_lds[lane] = lg;   // 0 for invalid rows
    }

    // ---- lane-parallel slot assignment over sorted ids (ballot prefix-sum) ----
    int lastact = (na > 0) ? active_ids[na - 1] : -2;   // LDS broadcast
    int previd = __shfl_up(myid, 1);
    if (lane == 0) previd = lastact;
    bool isnew = valid && (myid != previd);
    unsigned m1 = (unsigned)__ballot(isnew);
    int newd = __popc(m1 & 0xFFFFu);
    bool flushf = (na + newd > MAX_SLOTS);              // wave-uniform

    if (flushf) {
      // flush with OLD active_ids, reset window
      #pragma unroll
      for (int ct = 0; ct < NCT; ++ct) {
        #pragma unroll
        for (int rr = 0; rr < 8; ++rr) {
          int m = half * 8 + rr;     // C layout: VGPR rr -> M=rr (+8 for hi lanes)
          if (m < na)
            atomicAdd(&out[(long long)active_ids[m] * DCOLS + ct * 16 + lane16],
                      acc[ct][rr]);
          acc[ct][rr] = 0.0f;
        }
      }
      __syncthreads();
      na = 0;
      if (valid && lane == 0 && myid == lastact) isnew = true;  // id re-enters fresh window
    }

    unsigned m2 = (unsigned)__ballot(isnew);
    int pre  = __popc(m2 & ((2u << lane) - 1));   // inclusive prefix count (lanes<16)
    int slot = na + pre - 1;
    if (valid) {
      slot_lds[lane] = slot;
      if (isnew) active_ids[slot] = myid;
    }
    na += __popc(m2 & 0xFFFFu);
    __syncthreads();

    // ---- WMMA: D(16 slots x 16 cols) += A(16x4 one-hot*logit) x B(4x16 H) ----
    // A (32-bit 16x4): M=lane%16; lanes0-15 K=0,1 ; lanes16-31 K=2,3
    // B (4x16): N=lane%16; same K split across lane halves
    #pragma unroll
    for (int ks = 0; ks < 4; ++ks) {
      v2f afrag;
      #pragma unroll
      for (int j = 0; j < 2; ++j) {
        int r = ks * 4 + half * 2 + j;
        int s = slot_lds[r];
        afrag[j] = (s == lane16) ? logit_lds[r] : 0.0f;
      }
      #pragma unroll
      for (int ct = 0; ct < NCT; ++ct) {
        v2f bfrag;
        #pragma unroll
        for (int j = 0; j < 2; ++j) {
          int r = ks * 4 + half * 2 + j;
          bfrag[j] = hb[r * LDS_STRIDE + ct * 16 + lane16];
        }
        acc[ct] = __builtin_amdgcn_wmma_f32_16x16x4_f32(
            false, afrag, false, bfrag, (short)0, acc[ct], false, false);
      }
    }
    __syncthreads();     // next iteration's async copies overwrite the other buffer
  }

  // ---- final flush of remaining slots ----
  #pragma unroll
  for (int ct = 0; ct < NCT; ++ct) {
    #pragma unroll
    for (int rr = 0; rr < 8; ++rr) {
      int m = half * 8 + rr;
      if (m < na)
        atomicAdd(&out[(long long)active_ids[m] * DCOLS + ct * 16 + lane16],
                  acc[ct][rr]);
    }
  }
}

// Zero d_out (atomic-accumulated) and Z every launch — harness never re-zeros.
__global__ void zero_out_kernel(float* __restrict__ out, int n_out,
                                float* __restrict__ Z, int n_z) {
  int i = blockIdx.x * blockDim.x + threadIdx.x;
  if (i < n_out) out[i] = 0.0f;
  if (i < n_z)   Z[i]   = 0.0f;
}

// Epilogue: out[g,:] = S[g,:] / Z[g]  (0 for empty graphs)
__global__ void normalize_kernel(float* __restrict__ out,
                                 const float* __restrict__ Z, int n) {
  int i = blockIdx.x * blockDim.x + threadIdx.x;
  if (i < n) {
    float z = Z[i >> 7];     // D = 128
    float v = out[i];
    out[i] = (z != 0.0f) ? (v / z) : 0.0f;
  }
}

extern "C" void kernel_launch(void* const* d_in, const int* in_sizes, int n_in,
                              void* d_out, int out_size, void* d_ws, size_t ws_size,
                              hipStream_t stream) {
  const float* H     = (const float*)d_in[0];
  const int*   batch = (const int*)  d_in[1];
  const float* w     = (const float*)d_in[2];
  const float* b     = (const float*)d_in[3];
  float* out = (float*)d_out;
  float* Z   = (float*)d_ws;               // 4096 floats of scratch
  const int V = in_sizes[1];               // 1,000,000
  const int G = out_size / DCOLS;          // 4096

  zero_out_kernel<<<(out_size + 255) / 256, 256, 0, stream>>>(out, out_size, Z, G);

  const int nblocks = (V + CHUNK_ROWS - 1) / CHUNK_ROWS;   // one wave per 128 rows
  attn_pool_fused<<<nblocks, 32, 0, stream>>>(H, batch, w, b, out, Z, V);

  normalize_kernel<<<(out_size + 255) / 256, 256, 0, stream>>>(out, Z, out_size);
}